// SubTBLoss_67688684585508
// MI455X (gfx1250) — compile-verified
//
#include <hip/hip_runtime.h>
#include <hip/hip_bf16.h>

typedef __attribute__((ext_vector_type(2))) float v2f;
typedef __attribute__((ext_vector_type(8))) float v8f;

// log2(0.9)
#define LOG2_LAMBDA (-0.15200309344504997f)
// 0.9^16
#define LAMBDA_16 0.1853020188851841f
// band tiles per row-tile: covers j-i up to 16*64+15 = 1039 > 980 (f32 underflow bound of 0.9^L)
#define TSTEPS 65
#define SCAN_THREADS 1024

// ---------------------------------------------------------------------------
// Kernel 1: build d[k] = lfr[k] + cpb[k] - cpf[k], k = 0..T  (single workgroup)
// cpf[k] = sum log_pfs[0..k-1] (cpf[0]=0), same for cpb. lfr[T] = log_reward.
// ---------------------------------------------------------------------------
__global__ void __launch_bounds__(SCAN_THREADS)
subtb_build_d(const float* __restrict__ log_pfs,
              const float* __restrict__ log_pbs,
              const float* __restrict__ log_flows,
              const float* __restrict__ log_reward,
              float* __restrict__ d, int T) {
    __shared__ float spf[SCAN_THREADS];
    __shared__ float spb[SCAN_THREADS];
    const int tid = threadIdx.x;
    const int chunk = (T + SCAN_THREADS - 1) / SCAN_THREADS; // 8 for T=8192
    const int beg = tid * chunk;

    float sf = 0.f, sb = 0.f;
    for (int e = 0; e < chunk; ++e) {
        int k = beg + e;
        if (k < T) { sf += log_pfs[k]; sb += log_pbs[k]; }
    }
    spf[tid] = sf; spb[tid] = sb;
    __syncthreads();

    // Hillis-Steele inclusive scan over per-thread sums
    for (int off = 1; off < SCAN_THREADS; off <<= 1) {
        float af = 0.f, ab = 0.f;
        if (tid >= off) { af = spf[tid - off]; ab = spb[tid - off]; }
        __syncthreads();
        spf[tid] += af; spb[tid] += ab;
        __syncthreads();
    }

    float cf = (tid > 0) ? spf[tid - 1] : 0.f;  // cpf[beg]
    float cb = (tid > 0) ? spb[tid - 1] : 0.f;  // cpb[beg]
    for (int e = 0; e < chunk; ++e) {
        int k = beg + e;
        if (k < T) {
            d[k] = log_flows[k] + (cb - cf);     // uses prefix BEFORE k
            cf += log_pfs[k];
            cb += log_pbs[k];
        }
    }
    if (tid == SCAN_THREADS - 1) {
        d[T] = log_reward[0] + (spb[SCAN_THREADS - 1] - spf[SCAN_THREADS - 1]);
    }
}

// ---------------------------------------------------------------------------
// Kernel 2: banded weighted pairwise loss, weights baked into WMMA operands.
//   A row u   (16x4): lambda^{-u} * [x^2, x, 1, 0]      (zeroed if i0+u > T)
//   B col v(t)(4x16): lambda^{16t+v} * [1, -2y, y^2, 0] (zeroed if j0+v > T)
//   => D[u][v] = lambda^{j-i} (x_u - y_v)^2  directly from the matrix pipe.
// t=0 (diagonal tile, needs j>i element mask) is peeled; t=1..64 accumulate
// into the WMMA C operand (chained GEMM), so the inner loop is
// 1 load + ~6 VALU + 1 v_wmma_f32_16x16x4_f32.
// ---------------------------------------------------------------------------
__global__ void __launch_bounds__(256)
subtb_band_wmma(const float* __restrict__ d,
                float* __restrict__ partial_tot,
                int Np1) {
    const int wave = (int)((blockIdx.x * blockDim.x + threadIdx.x) >> 5);
    const int lane = (int)(threadIdx.x & 31);
    const int nRowTiles = (Np1 + 15) >> 4;
    if (wave >= nRowTiles) return;  // wave-uniform exit; EXEC all-ones below

    const int last = Np1 - 1;       // 8192
    const int i0 = wave << 4;
    const int Nn = lane & 15;       // column index v (B/D), also row index M for A fill
    const int half = lane >> 4;

    const float lamP = exp2f((float)Nn * LOG2_LAMBDA);    // lambda^{v}
    const float lamM = exp2f(-(float)Nn * LOG2_LAMBDA);   // lambda^{-u}, u = Nn

    // --- A (16x4 f32, 2 VGPRs): lanes 0-15 hold K=0,1 of row M=lane;
    //     lanes 16-31 hold K=2,3 of row M=lane-16. Row scaled by lambda^{-M}.
    int xi = i0 + Nn; if (xi > last) xi = last;
    const float x = d[xi];
    const bool rowvalid = (i0 + Nn) <= last;
    v2f A;
    A[0] = half ? lamM : (x * x * lamM);   // K=0: x^2*s | K=2: 1*s
    A[1] = half ? 0.0f : (x * lamM);       // K=1: x*s   | K=3: 0
    if (!rowvalid) { A[0] = 0.0f; A[1] = 0.0f; }

    float tot = 0.f;

    // ---- peeled t = 0 (diagonal tile): needs per-element j>i mask ----
    {
        const float gv = rowvalid ? lamP : 0.0f;  // j0==i0, col valid <=> row valid; f=1
        const float y = x;                        // same 16 values
        v2f B;
        B[0] = half ? (y * y * gv) : gv;          // K=0: 1*g  | K=2: y^2*g
        B[1] = half ? 0.0f : (-2.0f * y * gv);    // K=1: -2y*g| K=3: 0
        v8f c0 = {};
        c0 = __builtin_amdgcn_wmma_f32_16x16x4_f32(
            false, A, false, B, (short)0, c0, false, false);
#pragma unroll
        for (int r = 0; r < 8; ++r) {
            const int M = r + 8 * half;
            if (Nn > M) tot += c0[r];             // keep strictly-upper elements
        }
    }

    // ---- t = 1..TSTEPS-1: weights fully baked, accumulate in C ----
    float f = LAMBDA_16;                          // lambda^{16t}
    v8f cacc = {};
    for (int t = 1; t < TSTEPS; ++t) {
        const int j0 = i0 + (t << 4);
        const int yj = j0 + Nn;
        int yjc = yj; if (yjc > last) yjc = last;
        const float y = d[yjc];
        const float gv = (yj <= last) ? (f * lamP) : 0.0f;  // zero invalid columns
        v2f B;
        B[0] = half ? (y * y * gv) : gv;
        B[1] = half ? 0.0f : (-2.0f * y * gv);
        cacc = __builtin_amdgcn_wmma_f32_16x16x4_f32(
            false, A, false, B, (short)0, cacc, false, false);
        f *= LAMBDA_16;
    }
#pragma unroll
    for (int r = 0; r < 8; ++r) tot += cacc[r];

    // wave32 reduction
#pragma unroll
    for (int off = 16; off > 0; off >>= 1) {
        tot += __shfl_xor(tot, off, 32);
    }
    if (lane == 0) {
        partial_tot[wave] = tot;   // deterministic per-wave partials (no float atomics)
    }
}

// ---------------------------------------------------------------------------
// Kernel 3: reduce per-wave partials; wsum is data-independent:
//   wsum = sum_{L=1}^{N-1} (N-L) * lambda^L   (f32, underflows at L ~ 980
//   exactly like the reference's elementwise lambda**length in float32)
// ---------------------------------------------------------------------------
__global__ void __launch_bounds__(256)
subtb_finalize(const float* __restrict__ partial_tot,
               int n, int Np1, float* __restrict__ out) {
    __shared__ float st[256];
    __shared__ float sw[256];
    float t = 0.f, w = 0.f;
    for (int k = threadIdx.x; k < n; k += 256) t += partial_tot[k];
    for (int L = 1 + (int)threadIdx.x; L < Np1; L += 256)
        w += (float)(Np1 - L) * exp2f((float)L * LOG2_LAMBDA);
    st[threadIdx.x] = t; sw[threadIdx.x] = w;
    __syncthreads();
    for (int off = 128; off > 0; off >>= 1) {
        if ((int)threadIdx.x < off) {
            st[threadIdx.x] += st[threadIdx.x + off];
            sw[threadIdx.x] += sw[threadIdx.x + off];
        }
        __syncthreads();
    }
    if (threadIdx.x == 0) {
        const float tt = st[0], ww = sw[0];
        out[0] = (ww > 0.f) ? (tt / ww) : tt;
    }
}

extern "C" void kernel_launch(void* const* d_in, const int* in_sizes, int n_in,
                              void* d_out, int out_size, void* d_ws, size_t ws_size,
                              hipStream_t stream) {
    const float* log_pfs    = (const float*)d_in[0];
    const float* log_pbs    = (const float*)d_in[1];
    const float* log_flows  = (const float*)d_in[2];
    const float* log_reward = (const float*)d_in[3];
    const int T   = in_sizes[0];   // 8192
    const int Np1 = T + 1;         // 8193

    // workspace layout (floats): d[Np1] | partial_tot[nRowTiles]
    float* ws = (float*)d_ws;
    float* dvec = ws;
    const int nRowTiles = (Np1 + 15) >> 4;                 // 513
    float* partial_tot = ws + ((Np1 + 7) & ~7);

    subtb_build_d<<<1, SCAN_THREADS, 0, stream>>>(log_pfs, log_pbs, log_flows,
                                                  log_reward, dvec, T);

    const int wavesPerBlock = 8;  // 256 threads = 8 wave32
    const int blocks = (nRowTiles + wavesPerBlock - 1) / wavesPerBlock;
    subtb_band_wmma<<<blocks, 256, 0, stream>>>(dvec, partial_tot, Np1);

    subtb_finalize<<<1, 256, 0, stream>>>(partial_tot, nRowTiles, Np1,
                                          (float*)d_out);
}